// SpatialEmbLoss_4767413699036
// MI455X (gfx1250) — compile-verified
//
#include <hip/hip_runtime.h>
#include <hip/hip_bf16.h>

// ---------------------------------------------------------------------------
// SpatialEmbLoss forward for MI455X (gfx1250, wave32).
//   prediction: (2, 9, 768, 768) f32   [emb(2) | sigma(2) | seed(5)]
//   instances : (2, 768, 768) i32
//   labels    : (2, 768, 768) i32
//   out       : scalar f32 = mean of per-image losses
// Dist field evaluated via V_WMMA_F32_16X16X4_F32 (16 pixels x 15 instances,
// K=5 rank factorization of the anisotropic Gaussian argument).
// Lovasz hinge uses a stable 4x8-bit LSD radix sort with wave32 ballot
// multi-split ranking in the scatter.
// ---------------------------------------------------------------------------

#define HIMG 768
#define WIMG 768
#define NPIX (HIMG * WIMG)        // 589824
#define NCH 9
#define PP 15                     // 5 classes x 3 instance ids
#define NBK (NPIX / 256)          // 2304 blocks of 256
#define INV1023 (1.0f / 1023.0f)

typedef float v2f __attribute__((ext_vector_type(2)));
typedef float v8f __attribute__((ext_vector_type(8)));
typedef unsigned long long u64;
typedef unsigned int u32;

// ---- workspace byte offsets (total ~47.4 MB) ------------------------------
#define OFF_CNT      0u           // u32[30]
#define OFF_SIGSUM   128u         // f32[30][2]
#define OFF_SEEDBG   384u         // f32[2]
#define OFF_SEEDIN   512u         // f32[30]
#define OFF_VARSUM   640u         // f32[30]
#define OFF_LOV      768u         // f32[30]
#define OFF_MEDX     896u         // f32[30]
#define OFF_MEDY     1024u        // f32[30]
#define OFF_SM0      1152u        // f32[30]
#define OFF_SM1      1280u        // f32[30]
#define OFF_VALID    1408u        // u32[30]
#define OFF_COEFLO   1536u        // f32[2][16][4]
#define OFF_COEFHI   2048u        // f32[2][16][4]
#define OFF_HISTX    4096u        // u32[2][15][768]
#define OFF_HISTY    96256u       // u32[2][15][768]   end 188416
#define ZERO_WORDS   47104u       // 188416 / 4
#define OFF_ARGKEY   188416u      // u64[30]
#define OFF_COUNTS   188672u      // u32[256*NBK]      end 2547968
#define OFF_DIGTOT   2547968u     // u32[256]
#define OFF_DIGBASE  2548992u     // u32[256]
#define OFF_BLKCNT   2550016u     // u32[NBK]
#define OFF_BLKOFF   2559232u     // u32[NBK]          end 2568448
#define OFF_ERRS     2568448u     // f32[15][NPIX]     end 37957888
#define OFF_KEYA     37957888u    // u64[NPIX]
#define OFF_KEYB     42676480u    // u64[NPIX]         end 47395072

// ---- helpers ---------------------------------------------------------------
__device__ __forceinline__ float dev_sigmoid(float x) {
    return 1.0f / (1.0f + expf(-x));
}

__device__ __forceinline__ u32 inclScan256(u32 v, u32* s) {
    int t = threadIdx.x;
    s[t] = v; __syncthreads();
    for (int off = 1; off < 256; off <<= 1) {
        u32 x = (t >= off) ? s[t - off] : 0u;
        __syncthreads();
        s[t] += x;
        __syncthreads();
    }
    return s[t];
}

__device__ __forceinline__ float blockReduceF(float v, float* s) {
    int t = threadIdx.x;
    s[t] = v; __syncthreads();
    for (int off = 128; off > 0; off >>= 1) {
        if (t < off) s[t] += s[t + off];
        __syncthreads();
    }
    return s[0];
}

// ---- K0: zero accumulators, init argmin keys -------------------------------
__global__ void k_zero(u32* accWords, u64* argkey) {
    int i = blockIdx.x * blockDim.x + threadIdx.x;
    if (i < (int)ZERO_WORDS) accWords[i] = 0u;
    if (i < 30) argkey[i] = ~0ULL;
}

// ---- K1: per-pixel stats (seed_bg, counts, sigma sums, coord histograms) ---
__global__ void k_stats(const float* __restrict__ pred,
                        const int* __restrict__ inst,
                        const int* __restrict__ lab,
                        u32* cnt, float* sigsum, float* seedbg,
                        u32* histX, u32* histY) {
    __shared__ float red[256];
    __shared__ float s_sig0[PP], s_sig1[PP];
    __shared__ u32 s_cnt[PP];
    int t = threadIdx.x;
    if (t < PP) { s_sig0[t] = 0.f; s_sig1[t] = 0.f; s_cnt[t] = 0u; }
    __syncthreads();

    int gid = blockIdx.x * 256 + t;               // 0 .. 2*NPIX-1
    int img = gid / NPIX;
    int i = gid - img * NPIX;
    int r = i / WIMG, c = i - r * WIMG;
    const float* P = pred + (size_t)img * NCH * NPIX;
    int L = lab[(size_t)img * NPIX + i];
    int I = inst[(size_t)img * NPIX + i];

    float sbg = 0.f;
    #pragma unroll
    for (int k = 0; k < 5; ++k) {
        float v = dev_sigmoid(P[(size_t)(4 + k) * NPIX + i]);
        float v2 = v * v;
        sbg += v2;
        if (L == k + 1) sbg -= v2;                // only classes != label
    }
    float bs = blockReduceF(sbg, red);
    if (t == 0) atomicAdd(&seedbg[img], bs);
    __syncthreads();

    if (L >= 1 && L <= 5 && I >= 1 && I <= 3) {
        int p = (L - 1) * 3 + (I - 1);
        atomicAdd(&s_cnt[p], 1u);
        atomicAdd(&s_sig0[p], P[(size_t)2 * NPIX + i]);
        atomicAdd(&s_sig1[p], P[(size_t)3 * NPIX + i]);
        atomicAdd(&histX[(size_t)(img * PP + p) * WIMG + c], 1u);
        atomicAdd(&histY[(size_t)(img * PP + p) * WIMG + r], 1u);
    }
    __syncthreads();
    if (t < PP && (s_cnt[t] | __float_as_uint(s_sig0[t]) | __float_as_uint(s_sig1[t]))) {
        atomicAdd(&cnt[img * PP + t], s_cnt[t]);
        atomicAdd(&sigsum[(img * PP + t) * 2 + 0], s_sig0[t]);
        atomicAdd(&sigsum[(img * PP + t) * 2 + 1], s_sig1[t]);
    }
}

// ---- K2: medians from histograms + s_mean ----------------------------------
__global__ void k_centers(const u32* cnt, const float* sigsum,
                          const u32* histX, const u32* histY,
                          float* medx, float* medy, float* sm0, float* sm1,
                          u32* validArr) {
    int t = threadIdx.x;
    if (t >= 60) return;
    int img = t / 30;
    int rem = t - img * 30;
    int p = rem >> 1, dim = rem & 1;
    int ip = img * PP + p;
    u32 cn = cnt[ip];
    if (dim == 0) {
        validArr[ip] = (cn > 0) ? 1u : 0u;
        float cf = (float)(cn > 0 ? cn : 1u);
        sm0[ip] = sigsum[ip * 2 + 0] / cf;
        sm1[ip] = sigsum[ip * 2 + 1] / cf;
    }
    u32 mid = cn ? ((cn - 1u) >> 1) : 0u;
    const u32* h = (dim ? histY : histX) + (size_t)ip * WIMG;
    u32 cum = 0; float med = 0.f;
    for (int b = 0; b < WIMG; ++b) {
        cum += h[b];
        if (cum > mid) { med = (float)b * INV1023; break; }
    }
    (dim ? medy : medx)[ip] = med;
}

// ---- K3: argmin medoid pixel (packed atomicMin) + variance sums ------------
__global__ void k_argvar(const float* __restrict__ pred,
                         const int* __restrict__ inst,
                         const int* __restrict__ lab,
                         const float* medx, const float* medy,
                         const float* sm0, const float* sm1,
                         const u32* validArr, u64* argkey, float* varsum) {
    __shared__ float s_var[PP];
    int t = threadIdx.x;
    if (t < PP) s_var[t] = 0.f;
    __syncthreads();

    int gid = blockIdx.x * 256 + t;
    int img = gid / NPIX;
    int i = gid - img * NPIX;
    int r = i / WIMG, c = i - r * WIMG;
    int L = lab[(size_t)img * NPIX + i];
    int I = inst[(size_t)img * NPIX + i];
    if (L >= 1 && L <= 5 && I >= 1 && I <= 3) {
        int p = (L - 1) * 3 + (I - 1);
        int ip = img * PP + p;
        if (validArr[ip]) {
            float x = (float)c * INV1023, y = (float)r * INV1023;
            float dx = x - medx[ip], dy = y - medy[ip];
            float d = dx * dx + dy * dy;
            u64 key = ((u64)__float_as_uint(d) << 32) | (u32)i;  // ties -> min idx
            atomicMin(&argkey[ip], key);
            const float* P = pred + (size_t)img * NCH * NPIX;
            float g0 = P[(size_t)2 * NPIX + i] - sm0[ip];
            float g1 = P[(size_t)3 * NPIX + i] - sm1[ip];
            atomicAdd(&s_var[p], g0 * g0 + g1 * g1);
        }
    }
    __syncthreads();
    if (t < PP && s_var[t] != 0.f)
        atomicAdd(&varsum[img * PP + t], s_var[t]);
}

// ---- K4: finalize centers -> WMMA B-matrix coefficients --------------------
// arg(pixel,inst) = [ex^2, ex, ey^2, ey, 1] . [s0, -2 s0 cx, s1, -2 s1 cy, s0 cx^2 + s1 cy^2]
__global__ void k_finalize(const u64* argkey, const float* sm0, const float* sm1,
                           const u32* validArr, float* coefLo, float* coefHi) {
    int t = threadIdx.x;
    if (t >= 32) return;
    int img = t / 16, n = t & 15;
    float lo0 = 0.f, lo1 = 0.f, lo2 = 0.f, lo3 = 0.f, hi0 = 0.f;
    if (n < PP && validArr[img * PP + n]) {
        int ip = img * PP + n;
        u32 idx = (u32)(argkey[ip] & 0xFFFFFFFFull);
        float cx = (float)(idx % WIMG) * INV1023;
        float cy = (float)(idx / WIMG) * INV1023;
        float s0 = expf(sm0[ip] * 10.0f);
        float s1 = expf(sm1[ip] * 10.0f);
        lo0 = s0; lo1 = -2.0f * s0 * cx;
        lo2 = s1; lo3 = -2.0f * s1 * cy;
        hi0 = s0 * cx * cx + s1 * cy * cy;
    }
    float* cl = coefLo + (size_t)(img * 16 + n) * 4;
    float* ch = coefHi + (size_t)(img * 16 + n) * 4;
    cl[0] = lo0; cl[1] = lo1; cl[2] = lo2; cl[3] = lo3;
    ch[0] = hi0; ch[1] = 0.f; ch[2] = 0.f; ch[3] = 0.f;
}

// ---- K5: WMMA distance field + hinge errors + seed_in ----------------------
// One wave = 16 pixels x 16 instance columns; two K=4 f32 WMMA ops.
__global__ void k_dist(const float* __restrict__ P,     // image base of pred
                       const int* __restrict__ lab,     // image base
                       const int* __restrict__ inst,    // image base
                       const float* __restrict__ coefLo,// + img*64
                       const float* __restrict__ coefHi,
                       float* __restrict__ errs,        // [15][NPIX]
                       float* __restrict__ seedin) {    // &seedin[img*15]
    __shared__ float sacc[16];
    int t = threadIdx.x;
    if (t < 16) sacc[t] = 0.f;
    __syncthreads();

    int wave = t >> 5, lane = t & 31;
    int base = (blockIdx.x * 8 + wave) * 16;
    __builtin_prefetch(lab + base, 0, 0);
    __builtin_prefetch(inst + base, 0, 0);

    int m = lane & 15;
    bool hi = lane >= 16;
    int i = base + m;
    int r = i / WIMG, c = i - r * WIMG;
    float ex = tanhf(P[i]) + (float)c * INV1023;
    float ey = tanhf(P[(size_t)NPIX + i]) + (float)r * INV1023;

    // A tile 16x4 (K chunk 0): [ex^2, ex | ey^2, ey]; chunk 1: [1,0 | 0,0]
    float cc = hi ? ey : ex;
    v2f a0; a0.x = cc * cc; a0.y = cc;
    v2f a1; a1.x = hi ? 0.f : 1.f; a1.y = 0.f;

    // B tile 4x16: column n = lane&15, rows (0,1) on lanes<16, (2,3) on lanes>=16
    int n = m;
    int ko = hi ? 2 : 0;
    const float* bl = coefLo + (size_t)n * 4;
    const float* bh = coefHi + (size_t)n * 4;
    v2f b0; b0.x = bl[ko]; b0.y = bl[ko + 1];
    v2f b1; b1.x = bh[ko]; b1.y = bh[ko + 1];

    v8f acc = {};
    acc = __builtin_amdgcn_wmma_f32_16x16x4_f32(false, a0, false, b0,
                                                (short)0, acc, false, false);
    acc = __builtin_amdgcn_wmma_f32_16x16x4_f32(false, a1, false, b1,
                                                (short)0, acc, false, false);

    if (n < PP) {
        int clz = n / 3 + 1;
        int iid = n - (clz - 1) * 3 + 1;
        int half = hi ? 8 : 0;
        float local = 0.f;
        float ev[8];
        #pragma unroll
        for (int r8 = 0; r8 < 8; ++r8) {
            int ii = base + half + r8;
            float dist = expf(-acc[r8]);
            int Lv = lab[ii], Iv = inst[ii];
            bool gt = (Lv == clz) && (Iv == iid);
            ev[r8] = gt ? 2.0f * (1.0f - dist) : 2.0f * dist;
            if (gt) {
                float sd = dev_sigmoid(P[(size_t)(4 + clz - 1) * NPIX + ii]);
                float dd = sd - dist;
                local += dd * dd;
            }
        }
        // 8 consecutive floats -> two b128 stores (16B aligned by construction)
        float4* dstp = (float4*)(errs + (size_t)n * NPIX + base + half);
        dstp[0] = make_float4(ev[0], ev[1], ev[2], ev[3]);
        dstp[1] = make_float4(ev[4], ev[5], ev[6], ev[7]);
        if (local != 0.f) atomicAdd(&sacc[n], local);
    }
    __syncthreads();
    if (t < PP && sacc[t] != 0.f) atomicAdd(&seedin[t], sacc[t]);
}

// ---- K6: sort key generation (descending err, stable via index payload) ----
__global__ void k_keygen(const float* __restrict__ errs_p, u64* keyA,
                         const u32* vflag) {
    if (!*vflag) return;
    u32 gid = blockIdx.x * 256 + threadIdx.x;
    float e = errs_p[gid];
    u32 u = __float_as_uint(e);
    u = (u & 0x80000000u) ? ~u : (u | 0x80000000u);   // ascending-ordered
    u32 desc = ~u;                                     // descending-ordered
    keyA[gid] = ((u64)desc << 32) | gid;
}

// ---- radix sort: 8-bit digit histogram -------------------------------------
__global__ void k_rshist(const u64* __restrict__ src, u32* counts, int shift,
                         const u32* vflag) {
    if (!*vflag) return;
    __shared__ u32 h[256];
    int t = threadIdx.x;
    h[t] = 0u; __syncthreads();
    u64 key = src[blockIdx.x * 256 + t];
    u32 d8 = (u32)(key >> shift) & 255u;
    atomicAdd(&h[d8], 1u);
    __syncthreads();
    counts[(size_t)t * NBK + blockIdx.x] = h[t];
}

// per-digit exclusive scan across blocks (in place), plus digit totals
__global__ void k_rsscanblk(u32* counts, u32* digitTotal, const u32* vflag) {
    if (!*vflag) return;
    __shared__ u32 s[256];
    int t = threadIdx.x;
    int d = blockIdx.x;
    size_t base = (size_t)d * NBK;
    u32 carry = 0;
    for (int c0 = 0; c0 < NBK; c0 += 256) {
        u32 v = counts[base + c0 + t];
        u32 incl = inclScan256(v, s);
        u32 tot = s[255];
        counts[base + c0 + t] = incl - v + carry;
        carry += tot;
        __syncthreads();
    }
    if (t == 0) digitTotal[d] = carry;
}

__global__ void k_rsscandig(const u32* digitTotal, u32* digitBase,
                            const u32* vflag) {
    if (!*vflag) return;
    __shared__ u32 s[256];
    int t = threadIdx.x;
    u32 v = digitTotal[t];
    u32 incl = inclScan256(v, s);
    digitBase[t] = incl - v;
}

// stable scatter: wave32 ballot multi-split rank + per-wave digit histogram
__global__ void k_rsscatter(const u64* __restrict__ src, u64* __restrict__ dst,
                            const u32* counts, const u32* digitBase, int shift,
                            const u32* vflag) {
    if (!*vflag) return;
    __shared__ u32 waveHist[8][256];
    int t = threadIdx.x;
    #pragma unroll
    for (int w = 0; w < 8; ++w) waveHist[w][t] = 0u;
    __syncthreads();

    int wave = t >> 5, lane = t & 31;
    u64 key = src[blockIdx.x * 256 + t];
    u32 d8 = (u32)(key >> shift) & 255u;

    // lanes in this wave holding the same digit (8 ballots, one per digit bit)
    u32 match = 0xFFFFFFFFu;
    #pragma unroll
    for (int b = 0; b < 8; ++b) {
        u32 bal = (u32)__builtin_amdgcn_ballot_w32(((d8 >> b) & 1u) != 0u);
        match &= ((d8 >> b) & 1u) ? bal : ~bal;
    }
    u32 below = (1u << lane) - 1u;
    u32 inWaveRank = __popc(match & below);
    if (inWaveRank == 0) waveHist[wave][d8] = (u32)__popc(match);
    __syncthreads();

    u32 rank = inWaveRank;
    for (int w = 0; w < wave; ++w) rank += waveHist[w][d8];
    dst[digitBase[d8] + counts[(size_t)d8 * NBK + blockIdx.x] + rank] = key;
}

// ---- Lovasz: per-block positive counts -------------------------------------
__global__ void k_lvblocksum(const u64* __restrict__ keys,
                             const int* __restrict__ lab,
                             const int* __restrict__ inst,
                             int clz, int iid, u32* blkCnt, const u32* vflag) {
    if (!*vflag) return;
    __shared__ u32 s[256];
    int t = threadIdx.x;
    u32 idx = (u32)(keys[blockIdx.x * 256 + t] & 0xFFFFFFFFull);
    u32 g = (lab[idx] == clz && inst[idx] == iid) ? 1u : 0u;
    s[t] = g; __syncthreads();
    for (int off = 128; off > 0; off >>= 1) {
        if (t < off) s[t] += s[t + off];
        __syncthreads();
    }
    if (t == 0) blkCnt[blockIdx.x] = s[0];
}

__global__ void k_lvscan(const u32* blkCnt, u32* blkOff, const u32* vflag) {
    if (!*vflag) return;
    __shared__ u32 s[256];
    int t = threadIdx.x;
    u32 carry = 0;
    for (int c0 = 0; c0 < NBK; c0 += 256) {
        u32 v = blkCnt[c0 + t];
        u32 incl = inclScan256(v, s);
        u32 tot = s[255];
        blkOff[c0 + t] = incl - v + carry;
        carry += tot;
        __syncthreads();
    }
}

// Lovasz hinge: err_sorted . d(jaccard); jac_i needs only (i, prefix_i)
__global__ void k_lvfinal(const u64* __restrict__ keys,
                          const float* __restrict__ errs_p,
                          const int* __restrict__ lab,
                          const int* __restrict__ inst,
                          int clz, int iid, const u32* blkOff,
                          const u32* cntPtr, float* lovPtr, const u32* vflag) {
    if (!*vflag) return;
    __shared__ u32 s[256];
    __shared__ float red[256];
    int t = threadIdx.x;
    int gid = blockIdx.x * 256 + t;
    u32 idx = (u32)(keys[gid] & 0xFFFFFFFFull);
    u32 g = (lab[idx] == clz && inst[idx] == iid) ? 1u : 0u;
    u32 incl = inclScan256(g, s);
    u32 prefix = blkOff[blockIdx.x] + incl;          // inclusive cumsum(gt)
    float err = fmaxf(errs_p[idx], 0.0f);            // relu(errors_sorted)
    float gts = (float)(*cntPtr);
    float pf = (float)prefix;
    float i1 = (float)(gid + 1);
    float jac_i = 1.0f - (gts - pf) / (gts + i1 - pf);
    float term;
    if (gid == 0) {
        term = err * jac_i;
    } else {
        float pm1 = pf - (float)g;
        float jac_m = 1.0f - (gts - pm1) / (gts + (float)gid - pm1);
        term = err * (jac_i - jac_m);
    }
    float bs = blockReduceF(term, red);
    if (t == 0 && bs != 0.f) atomicAdd(lovPtr, bs);
}

// ---- K7: combine -----------------------------------------------------------
__global__ void k_combine(const u32* cnt, const u32* validArr, const float* lov,
                          const float* varsum, const float* seedin,
                          const float* seedbg, float* out) {
    if (threadIdx.x != 0 || blockIdx.x != 0) return;
    float total = 0.f;
    for (int img = 0; img < 2; ++img) {
        float vsum = 0.f, lovS = 0.f, varS = 0.f, seedS = 0.f;
        for (int p = 0; p < PP; ++p) {
            int ip = img * PP + p;
            if (validArr[ip]) {
                float cf = (float)(cnt[ip] > 0 ? cnt[ip] : 1u);
                vsum += 1.f;
                lovS += lov[ip];
                varS += varsum[ip] / (cf * 2.0f);    // / (cnts * NS)
                seedS += seedin[ip];                 // class weight == 1
            }
        }
        float objs = fmaxf(vsum, 1.0f);
        float loss = lovS / objs + 10.0f * varS / objs +
                     (seedbg[img] + seedS) / (float)NPIX;
        total += loss;
    }
    out[0] = 0.5f * total;
}

// ---------------------------------------------------------------------------
extern "C" void kernel_launch(void* const* d_in, const int* in_sizes, int n_in,
                              void* d_out, int out_size, void* d_ws, size_t ws_size,
                              hipStream_t stream) {
    const float* pred = (const float*)d_in[0];
    const int* inst = (const int*)d_in[1];
    const int* lab = (const int*)d_in[2];
    float* out = (float*)d_out;
    char* WS = (char*)d_ws;

    u32* cntP = (u32*)(WS + OFF_CNT);
    float* sigsumP = (float*)(WS + OFF_SIGSUM);
    float* seedbgP = (float*)(WS + OFF_SEEDBG);
    float* seedinP = (float*)(WS + OFF_SEEDIN);
    float* varsumP = (float*)(WS + OFF_VARSUM);
    float* lovP = (float*)(WS + OFF_LOV);
    float* medxP = (float*)(WS + OFF_MEDX);
    float* medyP = (float*)(WS + OFF_MEDY);
    float* sm0P = (float*)(WS + OFF_SM0);
    float* sm1P = (float*)(WS + OFF_SM1);
    u32* validP = (u32*)(WS + OFF_VALID);
    float* coefLoP = (float*)(WS + OFF_COEFLO);
    float* coefHiP = (float*)(WS + OFF_COEFHI);
    u32* histXP = (u32*)(WS + OFF_HISTX);
    u32* histYP = (u32*)(WS + OFF_HISTY);
    u64* argkeyP = (u64*)(WS + OFF_ARGKEY);
    u32* countsP = (u32*)(WS + OFF_COUNTS);
    u32* digTotP = (u32*)(WS + OFF_DIGTOT);
    u32* digBaseP = (u32*)(WS + OFF_DIGBASE);
    u32* blkCntP = (u32*)(WS + OFF_BLKCNT);
    u32* blkOffP = (u32*)(WS + OFF_BLKOFF);
    float* errsP = (float*)(WS + OFF_ERRS);
    u64* keyA = (u64*)(WS + OFF_KEYA);
    u64* keyB = (u64*)(WS + OFF_KEYB);

    // 1) zero accumulators / histograms, init argmin keys
    k_zero<<<(ZERO_WORDS + 255) / 256, 256, 0, stream>>>((u32*)WS, argkeyP);

    // 2) pixel statistics
    k_stats<<<2 * NBK, 256, 0, stream>>>(pred, inst, lab, cntP, sigsumP,
                                         seedbgP, histXP, histYP);

    // 3) exact medians + sigma means
    k_centers<<<1, 64, 0, stream>>>(cntP, sigsumP, histXP, histYP,
                                    medxP, medyP, sm0P, sm1P, validP);

    // 4) medoid argmin + variance sums
    k_argvar<<<2 * NBK, 256, 0, stream>>>(pred, inst, lab, medxP, medyP,
                                          sm0P, sm1P, validP, argkeyP, varsumP);

    // 5) instance Gaussian coefficients (WMMA B matrix)
    k_finalize<<<1, 32, 0, stream>>>(argkeyP, sm0P, sm1P, validP,
                                     coefLoP, coefHiP);

    for (int img = 0; img < 2; ++img) {
        const float* Pimg = pred + (size_t)img * NCH * NPIX;
        const int* labI = lab + (size_t)img * NPIX;
        const int* insI = inst + (size_t)img * NPIX;

        // 6) WMMA distance field -> hinge errors + seed_in
        k_dist<<<NBK * 2, 256, 0, stream>>>(Pimg, labI, insI,
                                            coefLoP + img * 64,
                                            coefHiP + img * 64,
                                            errsP, seedinP + img * PP);

        for (int p = 0; p < PP; ++p) {
            const u32* vf = validP + img * PP + p;
            int clz = p / 3 + 1, iid = p % 3 + 1;

            k_keygen<<<NBK, 256, 0, stream>>>(errsP + (size_t)p * NPIX, keyA, vf);

            // 7) 4-pass LSD radix sort on high-32 error key (stable)
            for (int pass = 0; pass < 4; ++pass) {
                const u64* src = (pass & 1) ? keyB : keyA;
                u64* dst = (pass & 1) ? keyA : keyB;
                int shift = 32 + 8 * pass;
                k_rshist<<<NBK, 256, 0, stream>>>(src, countsP, shift, vf);
                k_rsscanblk<<<256, 256, 0, stream>>>(countsP, digTotP, vf);
                k_rsscandig<<<1, 256, 0, stream>>>(digTotP, digBaseP, vf);
                k_rsscatter<<<NBK, 256, 0, stream>>>(src, dst, countsP,
                                                     digBaseP, shift, vf);
            }
            // sorted result in keyA (A->B->A->B->A)

            // 8) Lovasz hinge via 3-kernel prefix scan
            k_lvblocksum<<<NBK, 256, 0, stream>>>(keyA, labI, insI, clz, iid,
                                                  blkCntP, vf);
            k_lvscan<<<1, 256, 0, stream>>>(blkCntP, blkOffP, vf);
            k_lvfinal<<<NBK, 256, 0, stream>>>(keyA, errsP + (size_t)p * NPIX,
                                               labI, insI, clz, iid, blkOffP,
                                               cntP + img * PP + p,
                                               lovP + img * PP + p, vf);
        }
    }

    // 9) combine into scalar mean
    k_combine<<<1, 32, 0, stream>>>(cntP, validP, lovP, varsumP, seedinP,
                                    seedbgP, out);
}